// GradientGP_16862041604337
// MI455X (gfx1250) — compile-verified
//
#include <hip/hip_runtime.h>
#include <math.h>

// ---- problem constants (from reference) ----
#define NF 1024
#define DD 6
#define NG 512
#define NS 1024
#define MM 4096            // NF + NG*DD
#define NB 128             // Cholesky block
#define NRHS 1025          // NS + 1 (K_sx^T columns + y_train)
#define LDB 1056           // NRHS padded to multiple of 32
#define INVL2 0.25f        // 1/ell^2, ell=2
#define VARF 1.0f
#define NOISE_F (0.01f*0.01f + 1e-6f)
#define NOISE_G (0.01f*0.01f + 1e-6f)

typedef __attribute__((ext_vector_type(2))) float v2f;
typedef __attribute__((ext_vector_type(8))) float v8f;

__device__ __forceinline__ float rbf6(const float* __restrict__ a,
                                      const float* __restrict__ b) {
  float sq = 0.f;
#pragma unroll
  for (int d = 0; d < DD; ++d) { float t = a[d] - b[d]; sq += t * t; }
  return VARF * __expf(-0.5f * sq * INVL2);
}

__device__ __forceinline__ v8f wmma_f32(v2f a, v2f b, v8f c) {
  return __builtin_amdgcn_wmma_f32_16x16x4_f32(false, a, false, b,
                                               (short)0, c, false, false);
}

// ---------------- build K (MxM) ----------------
__global__ void k_build_K(const float* __restrict__ Xf,
                          const float* __restrict__ Xg,
                          float* __restrict__ K) {
  long long idx = (long long)blockIdx.x * blockDim.x + threadIdx.x;
  if (idx >= (long long)MM * MM) return;
  int i = (int)(idx / MM), j = (int)(idx % MM);
  float v;
  if (i < NF && j < NF) {
    v = rbf6(Xf + i * DD, Xf + j * DD);
  } else if (i < NF) {                       // K_fg
    int q = (j - NF) / DD, b = (j - NF) % DD;
    const float* xi = Xf + i * DD; const float* xq = Xg + q * DD;
    v = rbf6(xi, xq) * (xi[b] - xq[b]) * INVL2;
  } else if (j < NF) {                       // K_gf = K_fg^T
    int p = (i - NF) / DD, a = (i - NF) % DD;
    const float* xj = Xf + j * DD; const float* xp = Xg + p * DD;
    v = rbf6(xj, xp) * (xj[a] - xp[a]) * INVL2;
  } else {                                   // K_gg
    int p = (i - NF) / DD, a = (i - NF) % DD;
    int q = (j - NF) / DD, b = (j - NF) % DD;
    const float* xp = Xg + p * DD; const float* xq = Xg + q * DD;
    float da = xp[a] - xq[a], db = xp[b] - xq[b];
    float kk = rbf6(xp, xq) * INVL2;
    v = kk * ((a == b ? 1.f : 0.f) - da * db * INVL2);
  }
  if (i == j) v += (i < NF ? NOISE_F : NOISE_G);
  K[idx] = v;
}

// --------- build B = [K_sx^T | y | 0] (MxLDB) and K_sx (NsxM) ---------
__global__ void k_build_B(const float* __restrict__ Xs,
                          const float* __restrict__ Xf,
                          const float* __restrict__ Xg,
                          const float* __restrict__ Yf,
                          const float* __restrict__ Gg,
                          float* __restrict__ B,
                          float* __restrict__ Ksx) {
  long long idx = (long long)blockIdx.x * blockDim.x + threadIdx.x;
  if (idx >= (long long)MM * LDB) return;
  int m = (int)(idx / LDB), c = (int)(idx % LDB);
  float v = 0.f;
  if (c < NS) {
    const float* xs = Xs + c * DD;
    if (m < NF) {
      v = rbf6(xs, Xf + m * DD);
    } else {
      int p = (m - NF) / DD, b = (m - NF) % DD;
      const float* xp = Xg + p * DD;
      v = rbf6(xs, xp) * (xs[b] - xp[b]) * INVL2;
    }
    Ksx[(long long)c * MM + m] = v;
  } else if (c == NS) {
    v = (m < NF) ? Yf[m] : Gg[m - NF];
  }
  B[idx] = v;
}

// ---------------- Cholesky: diagonal block factor (1 workgroup) ----------------
__global__ void k_chol_diag(float* __restrict__ K, int k0) {
  extern __shared__ float s[];               // NB*NB floats (64 KB)
  int tid = threadIdx.x, nthr = blockDim.x;
  for (int t = tid; t < NB * NB; t += nthr)
    s[t] = K[(long long)(k0 + t / NB) * MM + k0 + t % NB];
  for (int j = 0; j < NB; ++j) {
    __syncthreads();
    float djj = sqrtf(s[j * NB + j]);
    float inv = 1.f / djj;
    __syncthreads();
    if (tid == 0) s[j * NB + j] = djj;
    for (int i = j + 1 + tid; i < NB; i += nthr) s[i * NB + j] *= inv;
    __syncthreads();
    int rem = NB - 1 - j;
    for (int t = tid; t < rem * rem; t += nthr) {
      int i = j + 1 + t / rem, l = j + 1 + t % rem;
      if (l <= i) s[i * NB + l] -= s[i * NB + j] * s[l * NB + j];
    }
  }
  __syncthreads();
  for (int t = tid; t < NB * NB; t += nthr)
    K[(long long)(k0 + t / NB) * MM + k0 + t % NB] = s[t];
}

// ---------------- Cholesky: panel TRSM (L21 = A21 * L11^-T) ----------------
__global__ void k_trsm(float* __restrict__ K, int k0) {
  extern __shared__ float xs[];              // NB * blockDim floats
  int tid = threadIdx.x;
  long long r = (long long)blockIdx.x * blockDim.x + tid + k0 + NB;
  if (r >= MM) return;
  const float* Ld = K + (long long)k0 * MM + k0;
  float* row = K + r * MM + k0;
  for (int j = 0; j < NB; ++j) {
    float ssum = row[j];
    for (int t = 0; t < j; ++t)
      ssum -= Ld[(long long)j * MM + t] * xs[t * blockDim.x + tid];
    float x = ssum / Ld[(long long)j * MM + j];
    xs[j * blockDim.x + tid] = x;
    row[j] = x;
  }
}

// ------- Cholesky: WMMA SYRK trailing update, 32x32 tile per wave -------
__global__ void __launch_bounds__(32, 1)
k_syrk(float* __restrict__ K, int k0) {
  if (blockIdx.y > blockIdx.x) return;       // lower-triangle tiles only (uniform)
  int lane = threadIdx.x;
  long long gi0 = k0 + NB + 32ll * blockIdx.x;
  long long gj0 = k0 + NB + 32ll * blockIdx.y;
  int mn = lane & 15;
  int kh = (lane >> 4) << 1;
  const float* A0 = K + (gi0 + mn) * MM + k0;
  const float* A1 = K + (gi0 + 16 + mn) * MM + k0;
  const float* B0 = K + (gj0 + mn) * MM + k0;
  const float* B1 = K + (gj0 + 16 + mn) * MM + k0;
  v8f c00 = {}, c01 = {}, c10 = {}, c11 = {};
#pragma unroll 8
  for (int kk = 0; kk < NB; kk += 4) {
    v2f a0, a1, b0, b1;
    a0.x = A0[kk + kh]; a0.y = A0[kk + kh + 1];
    a1.x = A1[kk + kh]; a1.y = A1[kk + kh + 1];
    b0.x = B0[kk + kh]; b0.y = B0[kk + kh + 1];
    b1.x = B1[kk + kh]; b1.y = B1[kk + kh + 1];
    c00 = wmma_f32(a0, b0, c00);
    c01 = wmma_f32(a0, b1, c01);
    c10 = wmma_f32(a1, b0, c10);
    c11 = wmma_f32(a1, b1, c11);
  }
  int col = lane & 15, rb = (lane >> 4) * 8;
#pragma unroll
  for (int r = 0; r < 8; ++r) {
    K[(gi0 + rb + r) * MM + gj0 + col]            -= c00[r];
    K[(gi0 + rb + r) * MM + gj0 + 16 + col]       -= c01[r];
    K[(gi0 + 16 + rb + r) * MM + gj0 + col]       -= c10[r];
    K[(gi0 + 16 + rb + r) * MM + gj0 + 16 + col]  -= c11[r];
  }
}

// ---------------- forward solve, diagonal block (per-column) ----------------
__global__ void k_fwd_diag(const float* __restrict__ K, float* __restrict__ B,
                           int k0) {
  extern __shared__ float xs[];              // NB * blockDim floats
  int tid = threadIdx.x;
  int c = blockIdx.x * blockDim.x + tid;
  if (c >= LDB) return;
  const float* Ld = K + (long long)k0 * MM + k0;
  for (int j = 0; j < NB; ++j) {
    float ssum = B[(long long)(k0 + j) * LDB + c];
    for (int t = 0; t < j; ++t)
      ssum -= Ld[(long long)j * MM + t] * xs[t * blockDim.x + tid];
    float x = ssum / Ld[(long long)j * MM + j];
    xs[j * blockDim.x + tid] = x;
    B[(long long)(k0 + j) * LDB + c] = x;
  }
}

// ------- forward solve, WMMA GEMM update: B2 -= L21 * W1 (32x32/wave) -------
__global__ void __launch_bounds__(32, 1)
k_fwd_update(const float* __restrict__ K, float* __restrict__ B, int k0) {
  int lane = threadIdx.x;
  long long r0 = k0 + NB + 32ll * blockIdx.x;
  long long c0 = 32ll * blockIdx.y;
  int mn = lane & 15;
  int kh = (lane >> 4) << 1;
  const float* A0 = K + (r0 + mn) * MM + k0;
  const float* A1 = K + (r0 + 16 + mn) * MM + k0;
  v8f c00 = {}, c01 = {}, c10 = {}, c11 = {};
#pragma unroll 8
  for (int kk = 0; kk < NB; kk += 4) {
    long long kb = k0 + kk + kh;
    v2f a0, a1, b0, b1;
    a0.x = A0[kk + kh]; a0.y = A0[kk + kh + 1];
    a1.x = A1[kk + kh]; a1.y = A1[kk + kh + 1];
    b0.x = B[kb * LDB + c0 + mn];       b0.y = B[(kb + 1) * LDB + c0 + mn];
    b1.x = B[kb * LDB + c0 + 16 + mn];  b1.y = B[(kb + 1) * LDB + c0 + 16 + mn];
    c00 = wmma_f32(a0, b0, c00);
    c01 = wmma_f32(a0, b1, c01);
    c10 = wmma_f32(a1, b0, c10);
    c11 = wmma_f32(a1, b1, c11);
  }
  int col = lane & 15, rb = (lane >> 4) * 8;
#pragma unroll
  for (int r = 0; r < 8; ++r) {
    B[(r0 + rb + r) * LDB + c0 + col]            -= c00[r];
    B[(r0 + rb + r) * LDB + c0 + 16 + col]       -= c01[r];
    B[(r0 + 16 + rb + r) * LDB + c0 + col]       -= c10[r];
    B[(r0 + 16 + rb + r) * LDB + c0 + 16 + col]  -= c11[r];
  }
}

// ---------------- backward solve, diagonal block (L^T) ----------------
__global__ void k_bwd_diag(const float* __restrict__ K, float* __restrict__ B,
                           int k0) {
  extern __shared__ float xs[];
  int tid = threadIdx.x;
  int c = blockIdx.x * blockDim.x + tid;
  if (c >= LDB) return;
  const float* Ld = K + (long long)k0 * MM + k0;
  for (int j = NB - 1; j >= 0; --j) {
    float ssum = B[(long long)(k0 + j) * LDB + c];
    for (int t = j + 1; t < NB; ++t)
      ssum -= Ld[(long long)t * MM + j] * xs[t * blockDim.x + tid];
    float x = ssum / Ld[(long long)j * MM + j];
    xs[j * blockDim.x + tid] = x;
    B[(long long)(k0 + j) * LDB + c] = x;
  }
}

// ------- backward solve, WMMA GEMM update: B_above -= L21^T * V1 -------
__global__ void __launch_bounds__(32, 1)
k_bwd_update(const float* __restrict__ K, float* __restrict__ B, int k0) {
  int lane = threadIdx.x;
  long long r0 = 32ll * blockIdx.x;
  long long c0 = 32ll * blockIdx.y;
  int mn = lane & 15;
  int kh = (lane >> 4) << 1;
  v8f c00 = {}, c01 = {}, c10 = {}, c11 = {};
#pragma unroll 8
  for (int kk = 0; kk < NB; kk += 4) {
    long long kb = k0 + kk + kh;
    v2f a0, a1, b0, b1;
    a0.x = K[kb * MM + r0 + mn];            // A(m,k) = L[k0+k][r0+m]
    a0.y = K[(kb + 1) * MM + r0 + mn];
    a1.x = K[kb * MM + r0 + 16 + mn];
    a1.y = K[(kb + 1) * MM + r0 + 16 + mn];
    b0.x = B[kb * LDB + c0 + mn];       b0.y = B[(kb + 1) * LDB + c0 + mn];
    b1.x = B[kb * LDB + c0 + 16 + mn];  b1.y = B[(kb + 1) * LDB + c0 + 16 + mn];
    c00 = wmma_f32(a0, b0, c00);
    c01 = wmma_f32(a0, b1, c01);
    c10 = wmma_f32(a1, b0, c10);
    c11 = wmma_f32(a1, b1, c11);
  }
  int col = lane & 15, rb = (lane >> 4) * 8;
#pragma unroll
  for (int r = 0; r < 8; ++r) {
    B[(r0 + rb + r) * LDB + c0 + col]            -= c00[r];
    B[(r0 + rb + r) * LDB + c0 + 16 + col]       -= c01[r];
    B[(r0 + 16 + rb + r) * LDB + c0 + col]       -= c10[r];
    B[(r0 + 16 + rb + r) * LDB + c0 + 16 + col]  -= c11[r];
  }
}

// ------- final: out = K_sx @ V (32x32/wave); mean & cov = K_ss - out -------
__global__ void __launch_bounds__(32, 1)
k_final(const float* __restrict__ Ksx, const float* __restrict__ V,
        const float* __restrict__ Xs, float* __restrict__ out) {
  int lane = threadIdx.x;
  long long s0 = 32ll * blockIdx.x;          // rows over Ns
  long long c0 = 32ll * blockIdx.y;          // cols over LDB
  int mn = lane & 15;
  int kh = (lane >> 4) << 1;
  const float* A0 = Ksx + (s0 + mn) * MM;
  const float* A1 = Ksx + (s0 + 16 + mn) * MM;
  v8f c00 = {}, c01 = {}, c10 = {}, c11 = {};
#pragma unroll 8
  for (int kk = 0; kk < MM; kk += 4) {
    long long kb = kk + kh;
    v2f a0, a1, b0, b1;
    a0.x = A0[kk + kh]; a0.y = A0[kk + kh + 1];
    a1.x = A1[kk + kh]; a1.y = A1[kk + kh + 1];
    b0.x = V[kb * LDB + c0 + mn];       b0.y = V[(kb + 1) * LDB + c0 + mn];
    b1.x = V[kb * LDB + c0 + 16 + mn];  b1.y = V[(kb + 1) * LDB + c0 + 16 + mn];
    c00 = wmma_f32(a0, b0, c00);
    c01 = wmma_f32(a0, b1, c01);
    c10 = wmma_f32(a1, b0, c10);
    c11 = wmma_f32(a1, b1, c11);
  }
  int col = lane & 15, rb = (lane >> 4) * 8;
#pragma unroll
  for (int ri = 0; ri < 2; ++ri) {
#pragma unroll
    for (int ci = 0; ci < 2; ++ci) {
      int ccol = (int)c0 + ci * 16 + col;
#pragma unroll
      for (int r = 0; r < 8; ++r) {
        int srow = (int)s0 + ri * 16 + rb + r;
        float acc = (ri == 0) ? (ci == 0 ? c00[r] : c01[r])
                              : (ci == 0 ? c10[r] : c11[r]);
        if (ccol < NS) {
          float kss = rbf6(Xs + srow * DD, Xs + ccol * DD);
          out[1024 + (long long)srow * NS + ccol] = kss - acc;   // cov
        } else if (ccol == NS) {
          out[srow] = acc;                                       // mean
        }
      }
    }
  }
}

extern "C" void kernel_launch(void* const* d_in, const int* in_sizes, int n_in,
                              void* d_out, int out_size, void* d_ws, size_t ws_size,
                              hipStream_t stream) {
  const float* Xf = (const float*)d_in[0];
  const float* Yf = (const float*)d_in[1];
  const float* Xg = (const float*)d_in[2];
  const float* Gg = (const float*)d_in[3];
  const float* Xs = (const float*)d_in[4];
  float* K   = (float*)d_ws;                          // M*M
  float* B   = K + (size_t)MM * MM;                   // M*LDB
  float* Ksx = B + (size_t)MM * LDB;                  // Ns*M
  float* out = (float*)d_out;

  const size_t ldsNB = (size_t)NB * NB * sizeof(float);     // 64 KB
  const size_t ldsCol = (size_t)NB * 128 * sizeof(float);   // 64 KB

  { long long tot = (long long)MM * MM;
    k_build_K<<<(unsigned)((tot + 255) / 256), 256, 0, stream>>>(Xf, Xg, K); }
  { long long tot = (long long)MM * LDB;
    k_build_B<<<(unsigned)((tot + 255) / 256), 256, 0, stream>>>(Xs, Xf, Xg, Yf, Gg, B, Ksx); }

  // Blocked Cholesky
  for (int k0 = 0; k0 < MM; k0 += NB) {
    k_chol_diag<<<1, 256, ldsNB, stream>>>(K, k0);
    int rows = MM - k0 - NB;
    if (rows > 0) {
      k_trsm<<<(rows + 127) / 128, 128, ldsCol, stream>>>(K, k0);
      dim3 g(rows / 32, rows / 32);
      k_syrk<<<g, 32, 0, stream>>>(K, k0);
    }
  }
  // Forward solve: L W = B
  for (int k0 = 0; k0 < MM; k0 += NB) {
    k_fwd_diag<<<(LDB + 127) / 128, 128, ldsCol, stream>>>(K, B, k0);
    int rows = MM - k0 - NB;
    if (rows > 0)
      k_fwd_update<<<dim3(rows / 32, LDB / 32), 32, 0, stream>>>(K, B, k0);
  }
  // Backward solve: L^T V = W
  for (int k0 = MM - NB; k0 >= 0; k0 -= NB) {
    k_bwd_diag<<<(LDB + 127) / 128, 128, ldsCol, stream>>>(K, B, k0);
    if (k0 > 0)
      k_bwd_update<<<dim3(k0 / 32, LDB / 32), 32, 0, stream>>>(K, B, k0);
  }
  // Final GEMM + mean/cov assembly
  k_final<<<dim3(NS / 32, LDB / 32), 32, 0, stream>>>(Ksx, B, Xs, out);
}